// QCQP_cvxpy_41326175322135
// MI455X (gfx1250) — compile-verified
//
#include <hip/hip_runtime.h>

// ---------------------------------------------------------------------------
// QCQP projected-gradient solver on MI455X (gfx1250), f32 WMMA path.
//   l <- proj_SOC( l - step*(l@P + q) ), 100 iters, N=24 (pad 32), BATCH=65536
// One wave32 handles 16 batch elements; all state register/LDS resident.
// Inner loop: 16x v_wmma_f32_16x16x4_f32 (two independent accumulator chains),
// 8 merged ds_store_2addr + 4 ds_load_2addr_b64 for the A-operand transpose,
// 24 ds_bpermute + v_sqrt/v_rcp for the SOC projection.
// ---------------------------------------------------------------------------

typedef __attribute__((ext_vector_type(2))) float v2f;   // WMMA f32 A/B operand (16x4 / 4x16)
typedef __attribute__((ext_vector_type(8))) float v8f;   // WMMA f32 C/D operand (16x16)

#define NCONES 8
#define NDIM 24
#define ITERS 100
#define BATCH 65536
#define WAVES_PER_BLOCK 8
#define LDS_STRIDE 34            // floats per l-tile row: even (8B-aligned b64 loads), padded

__device__ __forceinline__ float lane_gather(float v, int srcLane) {
  // wave32 cross-lane gather; bpermute index wraps mod 32, byte-addressed
  int r = __builtin_amdgcn_ds_bpermute(srcLane << 2, __builtin_bit_cast(int, v));
  return __builtin_bit_cast(float, r);
}

__device__ __forceinline__ float lane_gather_pre(float v, int byteIdx) {
  // variant with pre-shifted (loop-invariant) byte index
  int r = __builtin_amdgcn_ds_bpermute(byteIdx, __builtin_bit_cast(int, v));
  return __builtin_bit_cast(float, r);
}

__device__ __forceinline__ float wave_reduce_add(float x, int lane) {
#pragma unroll
  for (int off = 16; off >= 1; off >>= 1)
    x += lane_gather(x, lane ^ off);
  return x;
}

// ---------------------------------------------------------------------------
// Kernel 1: single-wave power iteration on the 24x24 SPD matrix P.
// Writes step = 1/lambda_max into ws[0]. (P = M M^T/N + 0.5 I is SPD, so
// power iteration converges to the top eigenvalue used by the reference.)
// ---------------------------------------------------------------------------
__global__ void __launch_bounds__(32)
power_iter_kernel(const float* __restrict__ P, float* __restrict__ step_out) {
  const int lane = threadIdx.x & 31;
  const bool act = lane < NDIM;
  const int rowi = act ? lane : (NDIM - 1);       // clamp: no OOB loads on idle lanes

  float row[NDIM];
#pragma unroll
  for (int j = 0; j < NDIM; ++j) {
    float p = P[rowi * NDIM + j];
    row[j] = act ? p : 0.0f;
  }

  float v = act ? 1.0f : 0.0f;
  for (int it = 0; it < 64; ++it) {
    float n2 = wave_reduce_add(v * v, lane);
    v *= rsqrtf(n2);
    float w = 0.0f;
#pragma unroll
    for (int j = 0; j < NDIM; ++j) w += row[j] * lane_gather(v, j);
    v = act ? w : 0.0f;
  }
  // Rayleigh quotient with normalized v
  float n2 = wave_reduce_add(v * v, lane);
  v *= rsqrtf(n2);
  float w = 0.0f;
#pragma unroll
  for (int j = 0; j < NDIM; ++j) w += row[j] * lane_gather(v, j);
  float lam = wave_reduce_add(v * w, lane);
  if (lane == 0) step_out[0] = 1.0f / lam;
}

// ---------------------------------------------------------------------------
// Kernel 2: main PGD loop. One wave per 16-element batch tile.
//   B-operands hold (-step * P) padded to 32x32, resident in VGPRs.
//   Accumulator seed C = l + (-step * q)  =>  WMMA chain yields l - step*(lP+q).
//   l lives in C/D layout registers; bounced through LDS to form A operands.
// ---------------------------------------------------------------------------
__global__ void __launch_bounds__(256)
soc_pgd_kernel(const float* __restrict__ P, const float* __restrict__ q,
               const float* __restrict__ step_p, float* __restrict__ out) {
  __shared__ float lds[WAVES_PER_BLOCK * 16 * LDS_STRIDE];

  const int tid = threadIdx.x;
  const int lane = tid & 31;
  const int wave = tid >> 5;
  const int li = lane & 15;        // column-within-tile / batch-row for A loads
  const int bhalf = lane >> 4;     // 0: lanes 0-15, 1: lanes 16-31
  const int half = lane & 16;      // lane-half base for bpermute (stay in batch half)
  const int batch_base = (blockIdx.x * WAVES_PER_BLOCK + wave) * 16;
  float* myLds = &lds[wave * 16 * LDS_STRIDE];

  const float nstep = -step_p[0];

  // ---- loop-invariant bpermute byte indices for the projection -------------
  // x-gather source register w: lanes 0-7 hold feats 16-23 (tile1),
  //                             lanes 8-15 hold feats 8-15 (tile0).
  // cone li: x0 at feature 8+2li -> lane (8+2li)&15 ; x1 at 9+2li.
  const int idx_x0 = (half + ((8 + 2 * li) & 15)) << 2;
  const int idx_x1 = (half + ((9 + 2 * li) & 15)) << 2;
  // scale scatter: l0-tail lanes (li>=8) need cone (li-8)>>1;
  //                l1-head lanes (li<8)  need cone 4+(li>>1).
  const int idx_sc = (half + ((li < 8) ? (4 + (li >> 1)) : ((li - 8) >> 1))) << 2;

  // ---- B operands: (-step)*P padded to 32x32, in WMMA 4x16 B layout --------
  // VGPR0: lanes0-15 row K=4k, lanes16-31 row K=4k+2; VGPR1: K=4k+1 / 4k+3.
  v2f B0[NCONES], B1[NCONES];
#pragma unroll
  for (int k = 0; k < NCONES; ++k) {
#pragma unroll
    for (int t = 0; t < 2; ++t) {
      int krow = 4 * k + 2 * bhalf + t;
      int krc = (krow < NDIM) ? krow : 0;                    // clamp OOB
      float p0 = nstep * P[krc * NDIM + li];                 // cols 0..15 always valid
      int c1 = (16 + li < NDIM) ? (16 + li) : 0;
      float p1 = nstep * P[krc * NDIM + c1];
      B0[k][t] = (krow < NDIM) ? p0 : 0.0f;
      B1[k][t] = (krow < NDIM && 16 + li < NDIM) ? p1 : 0.0f;
    }
  }

  // ---- qs = -step*q in C/D layout (lane=feature, VGPR r = batch r+8*bhalf) -
  float qs0[8], qs1[8];
#pragma unroll
  for (int r = 0; r < 8; ++r) {
    int b = batch_base + r + 8 * bhalf;
    qs0[r] = nstep * q[b * NDIM + li];
    int f1 = (li < NCONES) ? (16 + li) : 0;                  // clamp OOB
    float qv1 = nstep * q[b * NDIM + f1];
    qs1[r] = (li < NCONES) ? qv1 : 0.0f;
  }

  // ---- l state in C/D layout registers; l0 = 0 ----------------------------
  float l0[8], l1[8];
#pragma unroll
  for (int r = 0; r < 8; ++r) { l0[r] = 0.0f; l1[r] = 0.0f; }

  for (int it = 0; it < ITERS; ++it) {
    // 1) spill l to LDS row-major: row = batch, col = feature (in-wave, in-order DS)
#pragma unroll
    for (int r = 0; r < 8; ++r) {
      int rowb = r + 8 * bhalf;
      myLds[rowb * LDS_STRIDE + li] = l0[r];
      myLds[rowb * LDS_STRIDE + 16 + li] = l1[r];
    }

    // 2) A operands via ds_load_b64: lane li = batch row; the K/K+1 pair at
    //    offset 4k+2*bhalf lands exactly in the 16x4 f32 A register layout.
    v2f A[NCONES];
#pragma unroll
    for (int k = 0; k < NCONES; ++k)
      A[k] = *(const v2f*)&myLds[li * LDS_STRIDE + 4 * k + 2 * bhalf];

    // 3) seed C = l + (-step q); chain 16 WMMAs: D = A·(-step P) + C
    v8f c0, c1;
#pragma unroll
    for (int r = 0; r < 8; ++r) { c0[r] = l0[r] + qs0[r]; c1[r] = l1[r] + qs1[r]; }
#pragma unroll
    for (int k = 0; k < NCONES; ++k)
      c0 = __builtin_amdgcn_wmma_f32_16x16x4_f32(false, A[k], false, B0[k],
                                                 (short)0, c0, false, false);
#pragma unroll
    for (int k = 0; k < NCONES; ++k)
      c1 = __builtin_amdgcn_wmma_f32_16x16x4_f32(false, A[k], false, B1[k],
                                                 (short)0, c1, false, false);

    // 4) SOC projection. Cone i (i=0..7): head t = feature i, tail x = feats
    //    (8+2i, 9+2i). Head lanes (li<8, tile0) compute cone math; every lane
    //    then fetches the single scale value it needs via one bpermute.
#pragma unroll
    for (int r = 0; r < 8; ++r) {
      float v0 = c0[r];                     // feature li       of this batch row
      float v1 = c1[r];                     // feature 16+li
      float t = v0;                         // valid on head lanes li<8
      // merged x-source: lanes 0-7 -> tile1 (feats 16-23), lanes 8-15 -> tile0
      float w = (li < 8) ? v1 : v0;
      float x0 = lane_gather_pre(w, idx_x0);
      float x1 = lane_gather_pre(w, idx_x1);

      float n = __builtin_amdgcn_sqrtf(x0 * x0 + x1 * x1);
      bool inside = (n <= t);
      bool below = (n <= -t);
      float coef = 0.5f * (t + n);
      float tn = inside ? t : (below ? 0.0f : coef);
      float safen = (n > 0.0f) ? n : 1.0f;
      float sc = inside ? 1.0f : (below ? 0.0f : coef * __builtin_amdgcn_rcpf(safen));

      // one scale gather per lane: l0 tail lanes use cone (li-8)>>1,
      // l1 head lanes use cone 4+(li>>1)
      float s = lane_gather_pre(sc, idx_sc);
      l0[r] = (li < 8) ? tn : v0 * s;
      l1[r] = (li < 8) ? v1 * s : 0.0f;                       // keep padding at 0
    }
  }

  // ---- single global write of the result -----------------------------------
#pragma unroll
  for (int r = 0; r < 8; ++r) {
    int b = batch_base + r + 8 * bhalf;
    out[b * NDIM + li] = l0[r];
    if (li < NCONES) out[b * NDIM + 16 + li] = l1[r];
  }
}

// ---------------------------------------------------------------------------
extern "C" void kernel_launch(void* const* d_in, const int* in_sizes, int n_in,
                              void* d_out, int out_size, void* d_ws, size_t ws_size,
                              hipStream_t stream) {
  const float* P = (const float*)d_in[0];   // (1, 24, 24) f32
  const float* q = (const float*)d_in[1];   // (65536, 24, 1) f32
  float* out = (float*)d_out;               // (65536, 24) f32
  float* step_ws = (float*)d_ws;            // ws[0] = 1/lambda_max

  power_iter_kernel<<<1, 32, 0, stream>>>(P, step_ws);

  const int tiles = BATCH / 16;                               // 4096 waves
  const int blocks = tiles / WAVES_PER_BLOCK;                 // 512 blocks x 256 thr
  soc_pgd_kernel<<<blocks, 256, 0, stream>>>(P, q, step_ws, out);
}